// GATSingleHeadLayer_11914239279932
// MI455X (gfx1250) — compile-verified
//
#include <hip/hip_runtime.h>
#include <hip/hip_bf16.h>

typedef __attribute__((ext_vector_type(2))) float v2f;
typedef __attribute__((ext_vector_type(8))) float v8f;

#define INDIM  128
#define OUTDIM 64

// ---------------------------------------------------------------------------
// Kernel 1: z = h @ W  using V_WMMA_F32_16X16X4_F32.
// One wave -> 16x64 tile of z. 8 waves per workgroup.
// W staged in LDS k-pair-packed so each lane's B fragment {W[k][n],W[k+1][n]}
// is one aligned ds_load_b64 (no register shuffles feeding the WMMAs).
// ---------------------------------------------------------------------------
__global__ __launch_bounds__(256) void gat_gemm_z(const float* __restrict__ h,
                                                  const float* __restrict__ W,
                                                  float* __restrict__ z,
                                                  int nRows, int nTiles) {
    // Layout: Wlds[((k>>1)*OUTDIM + n)*2 + (k&1)]  -> float2 per (kpair, n)
    __shared__ float Wlds[INDIM * OUTDIM];  // 32 KB
    for (int i = threadIdx.x; i < INDIM * OUTDIM; i += 256) {
        int k = i >> 6;          // 0..127
        int n = i & 63;          // 0..63
        Wlds[(((k >> 1) * OUTDIM + n) << 1) | (k & 1)] = W[i];
    }
    __syncthreads();

    const int wave = threadIdx.x >> 5;
    const int lane = threadIdx.x & 31;
    const int tile = blockIdx.x * 8 + wave;
    if (tile >= nTiles) return;          // wave-uniform exit: EXEC stays all-1s

    const int rowBase = tile * 16;
    const int m  = lane & 15;            // row within tile (A) / col within ntile (B)
    const int hk = lane >> 4;            // K sub-pair selector

    int aRow = rowBase + m;
    if (aRow >= nRows) aRow = nRows - 1; // clamp loads; stores guarded below
    const float* __restrict__ hrow = h + (size_t)aRow * INDIM + hk * 2;

    // Per-lane base into packed W: kpair = k0/2 + hk, col = m (+16 per ntile)
    const v2f* __restrict__ wbase =
        (const v2f*)&Wlds[((hk * OUTDIM + m) << 1)];

    v8f acc0 = {}, acc1 = {}, acc2 = {}, acc3 = {};

    #pragma unroll 4
    for (int k0 = 0; k0 < INDIM; k0 += 4) {
        // A fragment: contiguous float2 {h[M][k0+2hk], h[M][k0+2hk+1]}
        v2f a = *(const v2f*)(hrow + k0);
        // B fragments: single b64 LDS loads, kpair stride = OUTDIM float2s
        const v2f* wp = wbase + (k0 >> 1) * OUTDIM;
        v2f b0 = wp[0];
        v2f b1 = wp[16];
        v2f b2 = wp[32];
        v2f b3 = wp[48];
        acc0 = __builtin_amdgcn_wmma_f32_16x16x4_f32(false, a, false, b0, (short)0, acc0, false, false);
        acc1 = __builtin_amdgcn_wmma_f32_16x16x4_f32(false, a, false, b1, (short)0, acc1, false, false);
        acc2 = __builtin_amdgcn_wmma_f32_16x16x4_f32(false, a, false, b2, (short)0, acc2, false, false);
        acc3 = __builtin_amdgcn_wmma_f32_16x16x4_f32(false, a, false, b3, (short)0, acc3, false, false);
    }

    // C/D layout: VGPR r -> row (r + 8*hk), col = m (+16 per ntile)
    #pragma unroll
    for (int r = 0; r < 8; ++r) {
        const int row = rowBase + r + hk * 8;
        if (row < nRows) {
            float* zp = z + (size_t)row * OUTDIM + m;
            zp[0]  = acc0[r];
            zp[16] = acc1[r];
            zp[32] = acc2[r];
            zp[48] = acc3[r];
        }
    }
}

// ---------------------------------------------------------------------------
// Kernel 2: per-node attention logit halves  s = z.Wa[:64], t = z.Wa[64:]
// ---------------------------------------------------------------------------
__global__ __launch_bounds__(256) void gat_st(const float* __restrict__ z,
                                              const float* __restrict__ Wa,
                                              float* __restrict__ s,
                                              float* __restrict__ t, int nRows) {
    int i = blockIdx.x * 256 + threadIdx.x;
    if (i >= nRows) return;
    const float* zr = z + (size_t)i * OUTDIM;
    float accS = 0.f, accT = 0.f;
    #pragma unroll
    for (int j = 0; j < OUTDIM; ++j) {
        float v = zr[j];
        accS = fmaf(v, Wa[j], accS);
        accT = fmaf(v, Wa[OUTDIM + j], accT);
    }
    s[i] = accS;
    t[i] = accT;
}

// ---------------------------------------------------------------------------
// Kernel 3: zero-init out (N*64), m (as uint bits, 0 == 0.0f), denom
// ---------------------------------------------------------------------------
__global__ __launch_bounds__(256) void gat_init(float* __restrict__ out,
                                                unsigned* __restrict__ m,
                                                float* __restrict__ denom,
                                                int nRows) {
    int i = blockIdx.x * 256 + threadIdx.x;
    if (i < nRows * OUTDIM) out[i] = 0.f;
    if (i < nRows) { m[i] = 0u; denom[i] = 0.f; }
}

// ---------------------------------------------------------------------------
// Kernel 4: e = relu(s[src]+t[dst]); segment max via uint atomicMax (e >= 0,
// so IEEE order == unsigned order and init 0 is the identity).
// ---------------------------------------------------------------------------
__global__ __launch_bounds__(256) void gat_edge_max(const int* __restrict__ src,
                                                    const int* __restrict__ dst,
                                                    const float* __restrict__ s,
                                                    const float* __restrict__ t,
                                                    float* __restrict__ ew,
                                                    unsigned* __restrict__ m, int nE) {
    int i = blockIdx.x * 256 + threadIdx.x;
    if (i >= nE) return;
    float e = s[src[i]] + t[dst[i]];
    e = fmaxf(e, 0.f);
    ew[i] = e;
    atomicMax(&m[dst[i]], __float_as_uint(e));
}

// ---------------------------------------------------------------------------
// Kernel 5: ex = exp(e - m[dst]); denom[dst] += ex (native f32 atomic add)
// ---------------------------------------------------------------------------
__global__ __launch_bounds__(256) void gat_edge_exp(const int* __restrict__ dst,
                                                    float* __restrict__ ew,
                                                    const unsigned* __restrict__ m,
                                                    float* __restrict__ denom, int nE) {
    int i = blockIdx.x * 256 + threadIdx.x;
    if (i >= nE) return;
    int d = dst[i];
    float ex = __expf(ew[i] - __uint_as_float(m[d]));
    ew[i] = ex;
    unsafeAtomicAdd(&denom[d], ex);
}

// ---------------------------------------------------------------------------
// Kernel 6: out[dst] += (ex/denom[dst]) * z[src] ; 64 lanes per edge,
// 4 edges per 256-thread block. Gathers of z[src] are 256B coalesced,
// atomic adds land row-coalesced on out[dst].
// ---------------------------------------------------------------------------
__global__ __launch_bounds__(256) void gat_scatter(const int* __restrict__ src,
                                                   const int* __restrict__ dst,
                                                   const float* __restrict__ ew,
                                                   const float* __restrict__ denom,
                                                   const float* __restrict__ z,
                                                   float* __restrict__ out, int nE) {
    int e = blockIdx.x * 4 + (threadIdx.x >> 6);
    int f = threadIdx.x & 63;
    if (e >= nE) return;
    int sIdx = src[e];
    int dIdx = dst[e];
    float alpha = ew[e] / denom[dIdx];
    unsafeAtomicAdd(&out[(size_t)dIdx * OUTDIM + f], alpha * z[(size_t)sIdx * OUTDIM + f]);
}

// ---------------------------------------------------------------------------
// Kernel 7: final ReLU in place on out
// ---------------------------------------------------------------------------
__global__ __launch_bounds__(256) void gat_relu(float* __restrict__ out, int n) {
    int i = blockIdx.x * 256 + threadIdx.x;
    if (i < n) out[i] = fmaxf(out[i], 0.f);
}

// ---------------------------------------------------------------------------
extern "C" void kernel_launch(void* const* d_in, const int* in_sizes, int n_in,
                              void* d_out, int out_size, void* d_ws, size_t ws_size,
                              hipStream_t stream) {
    const float* h   = (const float*)d_in[0];   // [N, 128]
    const float* W   = (const float*)d_in[1];   // [128, 64]
    const float* Wa  = (const float*)d_in[2];   // [128]
    const int*   src = (const int*)d_in[3];     // [E]
    const int*   dst = (const int*)d_in[4];     // [E]
    float* out = (float*)d_out;                 // [N, 64]

    const int N = in_sizes[0] / INDIM;
    const int E = in_sizes[3];

    // Workspace layout (floats): z[N*64] | s[N] | t[N] | m[N] | denom[N] | ew[E]
    float*    z     = (float*)d_ws;
    float*    s     = z + (size_t)N * OUTDIM;
    float*    t     = s + N;
    unsigned* m     = (unsigned*)(t + N);
    float*    denom = (float*)(m + N);
    float*    ew    = denom + N;

    const int nTiles = (N + 15) / 16;

    gat_gemm_z<<<(nTiles + 7) / 8, 256, 0, stream>>>(h, W, z, N, nTiles);
    gat_st<<<(N + 255) / 256, 256, 0, stream>>>(z, Wa, s, t, N);
    gat_init<<<(N * OUTDIM + 255) / 256, 256, 0, stream>>>(out, m, denom, N);
    gat_edge_max<<<(E + 255) / 256, 256, 0, stream>>>(src, dst, s, t, ew, m, E);
    gat_edge_exp<<<(E + 255) / 256, 256, 0, stream>>>(dst, ew, m, denom, E);
    gat_scatter<<<(E + 3) / 4, 256, 0, stream>>>(src, dst, ew, denom, z, out, E);
    gat_relu<<<(N * OUTDIM + 255) / 256, 256, 0, stream>>>(out, N * OUTDIM);
}